// VectorQuantizer_76115410420396
// MI455X (gfx1250) — compile-verified
//
#include <hip/hip_runtime.h>

typedef float v2f __attribute__((ext_vector_type(2)));
typedef float v4f __attribute__((ext_vector_type(4)));
typedef float v8f __attribute__((ext_vector_type(8)));
typedef unsigned int u32x4 __attribute__((ext_vector_type(4)));
typedef int i32x4 __attribute__((ext_vector_type(4)));
typedef int i32x8 __attribute__((ext_vector_type(8)));

#define N_E       8192
#define E_DIM     64
#define K_EXT     68           // 64 data + [-0.5*e2, 0, 0, 0]
#define BHW       32768        // 8 * 64 * 64
#define CH_STRIDE 4096         // H*W
#define B_STRIDE  (64 * 4096)  // C*H*W
#define TILE_BYTES (64 * K_EXT * 4)   // 17408 B per staged 64-code tile
#define EPS       1e-12f

// ---------------------------------------------------------------------------
// Kernel 1: L2-normalize codebook into extended rows e_ext[8192][68]:
// cols 0..63 = e_norm, col 64 = -0.5*||e_norm||^2, cols 65..67 = 0.
// With queries extended by a constant 1 at K=64, the WMMA chain computes
// score = dot(zn, en) - 0.5*e2 directly (argmax == reference argmin).
// ---------------------------------------------------------------------------
__global__ __launch_bounds__(256) void vq_norm_emb(const float* __restrict__ emb,
                                                   float* __restrict__ e_ext) {
  const int row  = blockIdx.x * 8 + (threadIdx.x >> 5);
  const int lane = threadIdx.x & 31;
  v2f v = *(const v2f*)(emb + row * E_DIM + 2 * lane);
  float ss = v.x * v.x + v.y * v.y;
#pragma unroll
  for (int m = 16; m >= 1; m >>= 1) ss += __shfl_xor(ss, m, 32);
  const float inv = 1.0f / fmaxf(sqrtf(ss), EPS);
  v.x *= inv; v.y *= inv;
  *(v2f*)(e_ext + row * K_EXT + 2 * lane) = v;
  if (lane == 0) {
    const v4f tail = {-0.5f * (ss * inv * inv), 0.f, 0.f, 0.f};
    *(v4f*)(e_ext + row * K_EXT + 64) = tail;
  }
}

// ---------------------------------------------------------------------------
// TDM: DMA one 64x68 fp32 tile (one row = one extended code) into LDS.
// Row stride 68 dwords is the conflict-free LDS layout, so no padding needed.
// ---------------------------------------------------------------------------
__device__ __forceinline__ void tdm_load_tile(unsigned lds_addr,
                                              const float* __restrict__ gptr) {
  const unsigned long long ga = (unsigned long long)(uintptr_t)gptr;
  u32x4 g0;
  g0.x = 1u;                                                // count=1 (load D#)
  g0.y = lds_addr;                                          // LDS byte address
  g0.z = (unsigned)(ga & 0xFFFFFFFFu);                      // global_addr[31:0]
  g0.w = (unsigned)((ga >> 32) & 0x01FFFFFFu) | (2u << 30); // addr[56:32] | type=2

  i32x8 g1;
  g1[0] = (2 << 16);        // data_size = 4B, no pad
  g1[1] = (K_EXT << 16);    // tensor_dim0 = 68 elements
  g1[2] = (64 << 16);       // tensor_dim1 = 64 rows
  g1[3] = (K_EXT << 16);    // tile_dim0   = 68 elements
  g1[4] = 64;               // tile_dim1   = 64 rows, tile_dim2 = 0
  g1[5] = K_EXT;            // tensor_dim0_stride = 68 elements
  g1[6] = 0;
  g1[7] = 0;

  const i32x4 gz = {0, 0, 0, 0};
#if defined(__clang_major__) && __clang_major__ >= 23
  const i32x8 gz8 = {0, 0, 0, 0, 0, 0, 0, 0};
  __builtin_amdgcn_tensor_load_to_lds(g0, g1, gz, gz, gz8, 0);
#else
  __builtin_amdgcn_tensor_load_to_lds(g0, g1, gz, gz, 0);
#endif
}

// ---------------------------------------------------------------------------
// Kernel 2: per-wave 16-query tile vs all 8192 codes, V_WMMA_F32_16X16X4_F32.
// Four N-subtile accumulator chains are manually interleaved so consecutive
// WMMAs are independent (dep distance 4) and pipeline through the matrix unit.
// Codebook tiles are double-buffered in LDS, fed by the Tensor Data Mover.
// ---------------------------------------------------------------------------
__global__ __launch_bounds__(256) void vq_main(const float* __restrict__ z,
                                               const float* __restrict__ e_ext,
                                               float* __restrict__ zq_out,
                                               float* __restrict__ idx_out) {
  __shared__ float s_e[2][64 * K_EXT];

  const int tid  = threadIdx.x;
  const int wave = tid >> 5;
  const int lane = tid & 31;
  const int hi   = lane >> 4;
  const int lr   = lane & 15;

  const unsigned lds_base = (unsigned)(unsigned long long)(uintptr_t)(void*)&s_e[0][0];

  // Prefetch tile 0 via TDM as early as possible (wave 0 drives the DMA).
  if (wave == 0) tdm_load_tile(lds_base, e_ext);

  // ---- load + normalize this wave's 16-row A tile; append K=64 constant 1 ----
  const int mbase = blockIdx.x * 128 + wave * 16;
  const int mrow  = mbase + lr;                 // A-matrix: M = lane % 16
  const float* zbase = z + (mrow >> 12) * B_STRIDE + (mrow & 4095);

  v2f a[17];
  float ss = 0.f;
#pragma unroll
  for (int f = 0; f < 16; ++f) {
    const int c0 = 4 * f + 2 * hi;              // K pair per ISA A layout
    const float x = zbase[c0 * CH_STRIDE];
    const float y = zbase[(c0 + 1) * CH_STRIDE];
    a[f].x = x; a[f].y = y;
    ss += x * x + y * y;
  }
  ss += __shfl_xor(ss, 16, 32);
  const float inv = 1.0f / fmaxf(sqrtf(ss), EPS);
#pragma unroll
  for (int f = 0; f < 16; ++f) { a[f].x *= inv; a[f].y *= inv; }
  a[16].x = (hi == 0) ? 1.0f : 0.0f;            // K=64 -> picks up -0.5*e2
  a[16].y = 0.0f;                               // K=65..67 are zero

  float best_s[8];
  int   best_n[8];
#pragma unroll
  for (int r = 0; r < 8; ++r) { best_s[r] = -__builtin_inff(); best_n[r] = 0; }

  for (int nb = 0; nb < N_E; nb += 64) {
    const int buf = (nb >> 6) & 1;
    __syncthreads();                            // consumers of buf^1 are done
    if (wave == 0) {
      if (nb + 64 < N_E) {
        tdm_load_tile(lds_base + (unsigned)((buf ^ 1) * TILE_BYTES),
                      e_ext + (nb + 64) * K_EXT);
        __builtin_amdgcn_s_wait_tensorcnt(1);   // tile nb has landed
      } else {
        __builtin_amdgcn_s_wait_tensorcnt(0);
      }
    }
    __syncthreads();                            // publish buf to all waves

    const float* eb[4];
#pragma unroll
    for (int s = 0; s < 4; ++s)
      eb[s] = &s_e[buf][(s * 16 + lr) * K_EXT + 2 * hi];

    v8f acc[4];
#pragma unroll
    for (int s = 0; s < 4; ++s) acc[s] = (v8f){0.f, 0.f, 0.f, 0.f, 0.f, 0.f, 0.f, 0.f};

#pragma unroll
    for (int f = 0; f < 17; ++f) {
#pragma unroll
      for (int s = 0; s < 4; ++s) {
        const v2f bfrag = *(const v2f*)(eb[s] + 4 * f);
        acc[s] = __builtin_amdgcn_wmma_f32_16x16x4_f32(
            false, a[f], false, bfrag, (short)0, acc[s], false, false);
      }
    }

#pragma unroll
    for (int s = 0; s < 4; ++s) {
      const int nidx = nb + s * 16 + lr;
#pragma unroll
      for (int r = 0; r < 8; ++r) {
        const bool take = acc[s][r] > best_s[r];   // strict: first index wins ties
        best_s[r] = take ? acc[s][r] : best_s[r];
        best_n[r] = take ? nidx      : best_n[r];
      }
    }
  }

  // ---- reduce argmax across the 16 lanes holding the same M rows ----
#pragma unroll
  for (int m = 1; m <= 8; m <<= 1) {
#pragma unroll
    for (int r = 0; r < 8; ++r) {
      const float os = __shfl_xor(best_s[r], m, 32);
      const int   on = __shfl_xor(best_n[r], m, 32);
      const bool take = (os > best_s[r]) || (os == best_s[r] && on < best_n[r]);
      best_s[r] = take ? os : best_s[r];
      best_n[r] = take ? on : best_n[r];
    }
  }

  // ---- write outputs: z_q = e_norm[idx] scattered to [B,C,H,W]; indices ----
#pragma unroll
  for (int r = 0; r < 8; ++r) {
    const int m   = mbase + r + 8 * hi;         // C/D layout: M = r + 8*(lane>=16)
    const int idx = best_n[r];
    if (lr == 0) idx_out[m] = (float)idx;
    const v4f ev = *(const v4f*)(e_ext + idx * K_EXT + 4 * lr);
    float* obase = zq_out + (m >> 12) * B_STRIDE + (m & 4095);
#pragma unroll
    for (int j = 0; j < 4; ++j)
      obase[(4 * lr + j) * CH_STRIDE] = ev[j];
  }
}

// ---------------------------------------------------------------------------
extern "C" void kernel_launch(void* const* d_in, const int* in_sizes, int n_in,
                              void* d_out, int out_size, void* d_ws, size_t ws_size,
                              hipStream_t stream) {
  const float* z   = (const float*)d_in[0];   // [8, 64, 64, 64]
  const float* emb = (const float*)d_in[1];   // [8192, 64]

  float* e_ext = (float*)d_ws;                // 8192*68 floats ~ 2.2 MB

  float* zq_out  = (float*)d_out;             // 2,097,152 floats
  float* idx_out = zq_out + 8 * 64 * 64 * 64; // 32,768 indices (as float)

  vq_norm_emb<<<N_E / 8, 256, 0, stream>>>(emb, e_ext);
  vq_main<<<BHW / 128, 256, 0, stream>>>(z, e_ext, zq_out, idx_out);
}